// CubeFlow_16741782519967
// MI455X (gfx1250) — compile-verified
//
#include <hip/hip_runtime.h>

typedef __attribute__((ext_vector_type(2))) float v2f;
typedef __attribute__((ext_vector_type(4))) float v4f;
typedef __attribute__((ext_vector_type(8))) float v8f;

// Single-instruction ReLU: v_med3_f32(x, 0, +inf) == clamp(x, 0, inf).
__device__ __forceinline__ float relu1(float x) {
  return __builtin_amdgcn_fmed3f(x, 0.0f, __builtin_inff());
}

// One 16-point WMMA tile: layer1 (8x v_wmma_f32_16x16x4_f32) + fused relu +
// layer2 (2 FMAs per accumulator register against W2 columns), then a
// distributed reduce-scatter butterfly: 15 shuffles total (vs 64 for a full
// per-value allreduce). Output index j = 2*r + o lands in lane t == j of each
// 16-lane half, which is exactly the coalesced probs memory layout.
__device__ __forceinline__ void mlp_tile(v2f a,
                                         const v2f (&bw)[8],
                                         const float (&b1v)[8],
                                         const float (&w2a)[8],
                                         const float (&w2b)[8],
                                         float b2lane,
                                         int lane,
                                         float* __restrict__ out_probs_tile) {
  float acc[16];
#pragma unroll
  for (int j = 0; j < 16; ++j) acc[j] = 0.0f;

#pragma unroll
  for (int n = 0; n < 8; ++n) {
    v8f c;
#pragma unroll
    for (int r = 0; r < 8; ++r) c[r] = b1v[n];  // bias depends only on N
    // D = A(16x4) * B(4x16) + C  -> h tile, M=point row, N=hidden unit
    c = __builtin_amdgcn_wmma_f32_16x16x4_f32(false, a, false, bw[n],
                                              (short)0, c, false, false);
#pragma unroll
    for (int r = 0; r < 8; ++r) {
      float h = relu1(c[r]);
      acc[2 * r]     = fmaf(h, w2a[n], acc[2 * r]);      // o = 0
      acc[2 * r + 1] = fmaf(h, w2b[n], acc[2 * r + 1]);  // o = 1
    }
  }

  // Reduce-scatter over each 16-lane half (masks 8,4,2,1 stay in-half).
  // Step with lane-mask m resolves output-index bit log2(m); lane keeps the
  // half of the array matching its own bit and receives the partner's fold.
  float s8[8];
#pragma unroll
  for (int i = 0; i < 8; ++i) {
    float lo = acc[i], hi = acc[i + 8];
    float send = (lane & 8) ? lo : hi;
    float recv = __shfl_xor(send, 8, 32);
    s8[i] = ((lane & 8) ? hi : lo) + recv;
  }
  float s4[4];
#pragma unroll
  for (int i = 0; i < 4; ++i) {
    float lo = s4[0] * 0.0f + s8[i], hi = s8[i + 4];  // (no-op; keep simple)
    lo = s8[i];
    float send = (lane & 4) ? lo : hi;
    float recv = __shfl_xor(send, 4, 32);
    s4[i] = ((lane & 4) ? hi : lo) + recv;
  }
  float s2[2];
#pragma unroll
  for (int i = 0; i < 2; ++i) {
    float lo = s2[0] * 0.0f + s4[i], hi = s4[i + 2];
    lo = s4[i];
    float send = (lane & 2) ? lo : hi;
    float recv = __shfl_xor(send, 2, 32);
    s2[i] = ((lane & 2) ? hi : lo) + recv;
  }
  {
    float lo = s2[0], hi = s2[1];
    float send = (lane & 1) ? lo : hi;
    float recv = __shfl_xor(send, 1, 32);
    float s1 = ((lane & 1) ? hi : lo) + recv;
    // lane t of each half now holds output j==t: probs[row t>>1][o = t&1]
    out_probs_tile[lane] = s1 + b2lane;
  }
}

__global__ void __launch_bounds__(256)
cubeflow_kernel(const float* __restrict__ input,   // (B,P,3)
                const float* __restrict__ latent,  // (B,1)
                const float* __restrict__ W1,      // (128,4)
                const float* __restrict__ b1,      // (128,)
                const float* __restrict__ W2,      // (2,128)
                const float* __restrict__ b2,      // (2,)
                float* __restrict__ out,
                int B, int P) {
  const int b = blockIdx.y;
  const int lane = threadIdx.x & 31;
  const int waveId = (blockIdx.x * blockDim.x + threadIdx.x) >> 5;
  const int nWaves = (gridDim.x * blockDim.x) >> 5;

  const float theta = latent[b] * 10.0f;  // uniform per batch row
  const float c = cosf(theta), s = sinf(theta);

  // ---- Preload weights into WMMA B layout (per wave, once) ----
  // B matrix 4x16: VGPR r = row K=r (lanes 0-15) / K=r+2 (lanes 16-31),
  // N = lane & 15. B[k][n] = W1[hidden n][k].
  const int col = lane & 15;
  const int khalf = (lane < 16) ? 0 : 2;
  v2f bw[8];
  float b1v[8], w2a[8], w2b[8];
#pragma unroll
  for (int n = 0; n < 8; ++n) {
    const int h = n * 16 + col;
    bw[n].x = W1[h * 4 + khalf];
    bw[n].y = W1[h * 4 + khalf + 1];
    b1v[n] = b1[h];
    w2a[n] = W2[h];          // W2[0][h]
    w2b[n] = W2[128 + h];    // W2[1][h]
  }
  const float b2lane = (lane & 1) ? b2[1] : b2[0];

  float* __restrict__ out_eval  = out;                      // B*P
  float* __restrict__ out_con   = out + (size_t)B * P;      // B*P*4
  float* __restrict__ out_probs = out + (size_t)B * P * 5;  // B*P*2

  const size_t batchBase = (size_t)b * P;

  // Each wave handles 32 points per step (two 16-point WMMA tiles).
  for (int base = waveId * 32; base < P; base += nWaves * 32) {
    const size_t gp = batchBase + (size_t)(base + lane);
    const float x0 = input[gp * 3 + 0];
    const float x1 = input[gp * 3 + 1];
    const float x2 = input[gp * 3 + 2];

    // ---- pointwise: con + cubeflow eval (z-axis rotation only) ----
    const float d0 = x0 - theta, d1 = x1, d2 = x2;
    const float f1 = 1.4f * d0 * d0 + 1.4f * d1 * d1 + 0.2f * d2 * d2 - 0.5f;
    const float e0 = d0, e1 = d1, e2 = d2 + 0.4f;
    const float c20 = c * e0 + s * e1;        // (R^T e)_0
    const float c21 = -s * e0 + c * e1;       // (R^T e)_1
    const float f2 = 3.8f * c20 * c20 + 0.6f * c21 * c21 + 3.8f * e2 * e2 - 0.5f;
    const float g0 = d0, g1 = d1, g2 = d2 - 0.6f;
    const float c30 = c * g0 - s * g1;        // (R g)_0
    const float c31 = s * g0 + c * g1;        // (R g)_1
    const float f3 = 0.35f * c30 * c30 + 2.8f * c31 * c31 + 2.8f * g2 * g2 - 0.5f;
    out_eval[gp] = fminf(f1, fminf(f2, f3));
    v4f conv = {x0, x1, x2, theta};
    *(v4f*)(out_con + gp * 4) = conv;  // coalesced 16B store

    // ---- build A operands for both tiles via one cross-half shuffle ----
    const float sx0 = __shfl_xor(x0, 16, 32);
    const float sx1 = __shfl_xor(x1, 16, 32);
    const float sx2 = __shfl_xor(x2, 16, 32);
    // A layout 16x4: lanes 0-15 hold (K0,K1)=(x0,x1) of M=lane,
    //                lanes 16-31 hold (K2,K3)=(x2,theta) of M=lane-16.
    v2f a0, a1;
    a0.x = (lane < 16) ? x0 : sx2;
    a0.y = (lane < 16) ? x1 : theta;
    a1.x = (lane < 16) ? sx0 : x2;
    a1.y = (lane < 16) ? sx1 : theta;

    mlp_tile(a0, bw, b1v, w2a, w2b, b2lane, lane,
             out_probs + (batchBase + base) * 2);
    mlp_tile(a1, bw, b1v, w2a, w2b, b2lane, lane,
             out_probs + (batchBase + base + 16) * 2);
  }
}

extern "C" void kernel_launch(void* const* d_in, const int* in_sizes, int n_in,
                              void* d_out, int out_size, void* d_ws, size_t ws_size,
                              hipStream_t stream) {
  const float* input  = (const float*)d_in[0];
  const float* latent = (const float*)d_in[1];
  const float* W1     = (const float*)d_in[2];
  const float* b1     = (const float*)d_in[3];
  const float* W2     = (const float*)d_in[4];
  const float* b2     = (const float*)d_in[5];

  const int B = in_sizes[1];               // latent is (B,1)
  const int P = in_sizes[0] / (B * 3);     // input is (B,P,3)

  dim3 block(256);                         // 8 wave32s
  dim3 grid(64, B);                        // 512 waves per batch row
  hipLaunchKernelGGL(cubeflow_kernel, grid, block, 0, stream,
                     input, latent, W1, b1, W2, b2, (float*)d_out, B, P);
}